// HybridPrecisionLinear_17111149707617
// MI455X (gfx1250) — compile-verified
//
#include <hip/hip_runtime.h>
#include <hip/hip_bf16.h>
#include <stdint.h>

// ---------------------------------------------------------------------------
// HybridPrecisionLinear for MI455X (gfx1250, wave32, WMMA)
//   out[64,8192] = x[64,8192] @ W_eff[8192,8192]^T + bias
//   W_eff = (|w| >= thr) ? w : ternary(w/scale)*scale
//   scale = max(mean|W|, 1e-5); thr = exact (n-k)-th smallest |w|, k=int(0.1 n)
// ---------------------------------------------------------------------------

#define IN_F   8192
#define OUT_F  8192
#define BATCH  64
#define N_W    67108864ull          // 8192*8192
#define KTH_IDX 60397977ull         // n - int(0.1*n) - 1
#define EPS_SCALE 1e-5f

// selection split: |w| bits = bits & 0x7fffffff (monotone key)
#define HI_BINS 8192                // bits[30:18]  (LDS histogram, 32 KB)
#define LO_BINS 262144              // bits[17:0]   (global histogram, 1 MB)

#define NBLK1 512                   // blocks for pass 1

typedef __attribute__((ext_vector_type(2))) float v2f;
typedef __attribute__((ext_vector_type(8))) float v8f;

// --------------------------- pass 1: sum|w| + hi-13 LDS histogram -----------
__global__ void __launch_bounds__(256)
k_abs_hist(const float* __restrict__ w,
           unsigned* __restrict__ hist1,
           float* __restrict__ partials) {
    __shared__ unsigned hsh[HI_BINS];
    __shared__ float red[256];
    for (int i = threadIdx.x; i < HI_BINS; i += 256) hsh[i] = 0u;
    __syncthreads();

    const uint4* w4 = (const uint4*)w;
    const int nthreads = gridDim.x * blockDim.x;
    const int tid = blockIdx.x * blockDim.x + threadIdx.x;
    const int n4 = (int)(N_W / 4);

    float s = 0.0f;
    for (int i = tid; i < n4; i += nthreads) {
        uint4 v = w4[i];
        unsigned a0 = v.x & 0x7fffffffu;
        unsigned a1 = v.y & 0x7fffffffu;
        unsigned a2 = v.z & 0x7fffffffu;
        unsigned a3 = v.w & 0x7fffffffu;
        s += (__uint_as_float(a0) + __uint_as_float(a1)) +
             (__uint_as_float(a2) + __uint_as_float(a3));
        atomicAdd(&hsh[a0 >> 18], 1u);
        atomicAdd(&hsh[a1 >> 18], 1u);
        atomicAdd(&hsh[a2 >> 18], 1u);
        atomicAdd(&hsh[a3 >> 18], 1u);
    }
    red[threadIdx.x] = s;
    __syncthreads();
    for (int off = 128; off > 0; off >>= 1) {
        if ((int)threadIdx.x < off) red[threadIdx.x] += red[threadIdx.x + off];
        __syncthreads();
    }
    if (threadIdx.x == 0) partials[blockIdx.x] = red[0];
    // flush sparse LDS histogram (hist atomics all precede the barriers above)
    for (int i = threadIdx.x; i < HI_BINS; i += 256) {
        unsigned c = hsh[i];
        if (c) atomicAdd(&hist1[i], c);
    }
}

// --------------------------- pass 2: scale + parallel hi rank-select --------
__global__ void __launch_bounds__(256)
k_select_hi(const float* __restrict__ partials,
            const unsigned* __restrict__ hist1,
            float* __restrict__ scalars,       // [0]=scale [1]=thr
            unsigned* __restrict__ uscal) {    // [0]=bin [1]=rank-in-bin
    __shared__ float red[256];
    __shared__ unsigned long long part[256];
    const int tid = threadIdx.x;

    float s = 0.0f;
    for (int i = tid; i < NBLK1; i += 256) s += partials[i];
    red[tid] = s;
    __syncthreads();
    for (int off = 128; off > 0; off >>= 1) {
        if (tid < off) red[tid] += red[tid + off];
        __syncthreads();
    }
    if (tid == 0) scalars[0] = fmaxf(red[0] / (float)N_W, EPS_SCALE);

    // rank select over HI_BINS: per-thread 32-bin range
    const int per = HI_BINS / 256;             // 32
    const int b0 = tid * per;
    unsigned long long loc = 0;
    for (int j = 0; j < per; ++j) loc += (unsigned long long)hist1[b0 + j];
    part[tid] = loc;
    __syncthreads();
    if (tid == 0) {                            // exclusive scan of 256 entries
        unsigned long long c = 0;
        for (int i = 0; i < 256; ++i) { unsigned long long t = part[i]; part[i] = c; c += t; }
    }
    __syncthreads();
    unsigned long long cum = part[tid];
    for (int j = 0; j < per; ++j) {
        unsigned long long c = (unsigned long long)hist1[b0 + j];
        if (cum <= KTH_IDX && KTH_IDX < cum + c) {
            uscal[0] = (unsigned)(b0 + j);
            uscal[1] = (unsigned)(KTH_IDX - cum);
        }
        cum += c;
    }
}

// --------------------------- pass 3: lo-18 histogram in selected bin --------
__global__ void __launch_bounds__(256)
k_hist_lo(const float* __restrict__ w,
          unsigned* __restrict__ hist2,
          const unsigned* __restrict__ uscal) {
    const unsigned bin = uscal[0];
    const uint4* w4 = (const uint4*)w;
    const int nthreads = gridDim.x * blockDim.x;
    const int tid = blockIdx.x * blockDim.x + threadIdx.x;
    const int n4 = (int)(N_W / 4);
    for (int i = tid; i < n4; i += nthreads) {
        uint4 v = w4[i];
        unsigned a0 = v.x & 0x7fffffffu;
        unsigned a1 = v.y & 0x7fffffffu;
        unsigned a2 = v.z & 0x7fffffffu;
        unsigned a3 = v.w & 0x7fffffffu;
        if ((a0 >> 18) == bin) atomicAdd(&hist2[a0 & 0x3ffffu], 1u);
        if ((a1 >> 18) == bin) atomicAdd(&hist2[a1 & 0x3ffffu], 1u);
        if ((a2 >> 18) == bin) atomicAdd(&hist2[a2 & 0x3ffffu], 1u);
        if ((a3 >> 18) == bin) atomicAdd(&hist2[a3 & 0x3ffffu], 1u);
    }
}

// --------------------------- pass 4: parallel lo rank-select -> thr ---------
__global__ void __launch_bounds__(256)
k_select_lo(const unsigned* __restrict__ hist2,
            const unsigned* __restrict__ uscal,
            float* __restrict__ scalars) {
    __shared__ unsigned long long part[256];
    const int tid = threadIdx.x;
    const unsigned bin  = uscal[0];
    const unsigned long long rank = (unsigned long long)uscal[1];

    const int per = LO_BINS / 256;             // 1024
    const int b0 = tid * per;
    unsigned long long loc = 0;
    for (int j = 0; j < per; ++j) loc += (unsigned long long)hist2[b0 + j];
    part[tid] = loc;
    __syncthreads();
    if (tid == 0) {
        unsigned long long c = 0;
        for (int i = 0; i < 256; ++i) { unsigned long long t = part[i]; part[i] = c; c += t; }
    }
    __syncthreads();
    unsigned long long cum = part[tid];
    for (int j = 0; j < per; ++j) {
        unsigned long long c = (unsigned long long)hist2[b0 + j];
        if (cum <= rank && rank < cum + c) {
            scalars[1] = __uint_as_float((bin << 18) | (unsigned)(b0 + j));
        }
        cum += c;
    }
}

// --------------------------- pass 5: WMMA GEMM, LDS-staged quantized B ------
__device__ __forceinline__ float quant_eff(float w, float scale, float half_s, float thr) {
    float a = fabsf(w);
    float t = (a > half_s) ? copysignf(scale, w) : 0.0f;   // ternary * scale
    return (a >= thr) ? w : t;                             // keep criticals fp32
}

#define KC      128                 // K-chunk in floats
#define KSTEPS  (KC / 4)            // 32 wmma steps per chunk
#define NCHUNK  (IN_F / KC)         // 64
#define BSTRIDE 66                  // LDS floats per step (pad 64->66: conflict-free)

__global__ void __launch_bounds__(128)
k_gemm(const float* __restrict__ x,
       const float* __restrict__ w,
       const float* __restrict__ bias,
       const float* __restrict__ scalars,
       float* __restrict__ out) {
    // quantized B fragments in wmma lane order: bq[t*BSTRIDE + 2*lane + {0,1}]
    __shared__ float bq[KSTEPS * BSTRIDE];

    const float scale  = scalars[0];
    const float thr    = scalars[1];
    const float half_s = 0.5f * scale;

    const int tidx  = threadIdx.x;
    const int lane  = tidx & 31;
    const int wave  = tidx >> 5;             // 0..3
    const int lm    = lane & 15;
    const int hi    = lane >> 4;
    const int obase = blockIdx.x * 16;

    // ---- producer mapping: wave w owns rows 4w..4w+3, lane = wmma step t ----
    const int pt    = lane;                  // step 0..31
    const int prow0 = wave * 4;
    const float4* wsrc0 = (const float4*)(w + (size_t)(obase + prow0 + 0) * IN_F);
    const float4* wsrc1 = (const float4*)(w + (size_t)(obase + prow0 + 1) * IN_F);
    const float4* wsrc2 = (const float4*)(w + (size_t)(obase + prow0 + 2) * IN_F);
    const float4* wsrc3 = (const float4*)(w + (size_t)(obase + prow0 + 3) * IN_F);

    // ---- consumer A side ----
    const float* xrow = x + (size_t)(wave * 16 + lm) * IN_F;

    v8f acc = {0.f, 0.f, 0.f, 0.f, 0.f, 0.f, 0.f, 0.f};

    float4 r0 = wsrc0[pt];                   // preload chunk 0
    float4 r1 = wsrc1[pt];
    float4 r2 = wsrc2[pt];
    float4 r3 = wsrc3[pt];

    for (int c = 0; c < NCHUNK; ++c) {
        // quantize once, scatter into wmma B-fragment order
        {
            float* d0 = &bq[pt * BSTRIDE + 2 * (prow0 + 0)];
            float* d1 = &bq[pt * BSTRIDE + 2 * (prow0 + 1)];
            float* d2 = &bq[pt * BSTRIDE + 2 * (prow0 + 2)];
            float* d3 = &bq[pt * BSTRIDE + 2 * (prow0 + 3)];
            d0[0] = quant_eff(r0.x, scale, half_s, thr);
            d0[1] = quant_eff(r0.y, scale, half_s, thr);
            d0[32] = quant_eff(r0.z, scale, half_s, thr);
            d0[33] = quant_eff(r0.w, scale, half_s, thr);
            d1[0] = quant_eff(r1.x, scale, half_s, thr);
            d1[1] = quant_eff(r1.y, scale, half_s, thr);
            d1[32] = quant_eff(r1.z, scale, half_s, thr);
            d1[33] = quant_eff(r1.w, scale, half_s, thr);
            d2[0] = quant_eff(r2.x, scale, half_s, thr);
            d2[1] = quant_eff(r2.y, scale, half_s, thr);
            d2[32] = quant_eff(r2.z, scale, half_s, thr);
            d2[33] = quant_eff(r2.w, scale, half_s, thr);
            d3[0] = quant_eff(r3.x, scale, half_s, thr);
            d3[1] = quant_eff(r3.y, scale, half_s, thr);
            d3[32] = quant_eff(r3.z, scale, half_s, thr);
            d3[33] = quant_eff(r3.w, scale, half_s, thr);
        }
        __syncthreads();

        // next chunk's weight loads in flight during the WMMA phase
        if (c + 1 < NCHUNK) {
            const int base = (c + 1) * KSTEPS + pt;
            r0 = wsrc0[base];
            r1 = wsrc1[base];
            r2 = wsrc2[base];
            r3 = wsrc3[base];
        }
        // deepen the HBM pipeline: prefetch 2 chunks ahead into caches
        if (c + 2 < NCHUNK) {
            const int pb = (c + 2) * KSTEPS + pt;
            __builtin_prefetch((const void*)(wsrc0 + pb), 0, 0);
            __builtin_prefetch((const void*)(wsrc1 + pb), 0, 0);
            __builtin_prefetch((const void*)(wsrc2 + pb), 0, 0);
            __builtin_prefetch((const void*)(wsrc3 + pb), 0, 0);
        }

        // consume: 32 WMMAs, B from LDS (conflict-free), A from global (L2-resident)
        const float* xk = xrow + c * KC + 2 * hi;
#pragma unroll 8
        for (int t = 0; t < KSTEPS; ++t) {
            float2 av = *(const float2*)(xk + 4 * t);
            float2 bv = *(const float2*)(&bq[t * BSTRIDE + 2 * lane]);
            v2f a; a.x = av.x; a.y = av.y;
            v2f b; b.x = bv.x; b.y = bv.y;
            acc = __builtin_amdgcn_wmma_f32_16x16x4_f32(
                      false, a, false, b, (short)0, acc, false, false);
        }
        __syncthreads();
    }

    const float bv = bias[obase + lm];
    // C/D layout: VGPR j -> M=j (lanes 0-15) / M=j+8 (lanes 16-31); N = lm
#pragma unroll
    for (int j = 0; j < 8; ++j) {
        const int m = wave * 16 + j + 8 * hi;
        out[(size_t)m * OUT_F + obase + lm] = acc[j] + bv;
    }
}

// ---------------------------------------------------------------------------
extern "C" void kernel_launch(void* const* d_in, const int* in_sizes, int n_in,
                              void* d_out, int out_size, void* d_ws, size_t ws_size,
                              hipStream_t stream) {
    const float* x    = (const float*)d_in[0];
    const float* wgt  = (const float*)d_in[1];
    const float* bias = (const float*)d_in[2];
    float* out = (float*)d_out;

    uint8_t* ws = (uint8_t*)d_ws;
    unsigned* hist1    = (unsigned*)ws;                                  // 32 KB
    unsigned* hist2    = (unsigned*)(ws + HI_BINS * 4u);                 // 1 MB
    float*    partials = (float*)   (ws + HI_BINS * 4u + LO_BINS * 4u);  // 2 KB
    float*    scalars  = (float*)   (ws + HI_BINS * 4u + LO_BINS * 4u + NBLK1 * 4u);
    unsigned* uscal    = (unsigned*)(scalars + 2);                       // bin, rank

    const size_t zero_bytes = (size_t)HI_BINS * 4u + (size_t)LO_BINS * 4u
                            + (size_t)NBLK1 * 4u + 16u;
    hipMemsetAsync(ws, 0, zero_bytes, stream);

    k_abs_hist <<<NBLK1, 256, 0, stream>>>(wgt, hist1, partials);
    k_select_hi<<<1,     256, 0, stream>>>(partials, hist1, scalars, uscal);
    k_hist_lo  <<<1024,  256, 0, stream>>>(wgt, hist2, uscal);
    k_select_lo<<<1,     256, 0, stream>>>(hist2, uscal, scalars);
    k_gemm     <<<OUT_F / 16, 128, 0, stream>>>(x, wgt, bias, scalars, out);
}